// EPropLSTM_87823491269161
// MI455X (gfx1250) — compile-verified
//
#include <hip/hip_runtime.h>

typedef __attribute__((ext_vector_type(2))) float v2f;
typedef __attribute__((ext_vector_type(8))) float v8f;

#define S_  128
#define B_  8
#define I_  256
#define H_  256
#define H4_ 1024
#define K_  (S_ * B_)          // 1024 = flattened (t, b) contraction dim
#define SBH (S_ * B_ * H_)     // 262144

// GEMM block tiling
#define BM  128                // output rows per block
#define BN  64                 // output cols per block
#define KC  32                 // K-chunk staged in LDS
#define PAD 34                 // padded floats per LDS row (even -> 8B-aligned b64)
#define NCH (K_ / KC)          // 32 chunks
#define ASYNC_PER_CHUNK 24     // 16 A + 8 B async issues per wave per chunk

__device__ __forceinline__ float sigmoidf_(float x) {
    return 1.0f / (1.0f + __expf(-x));
}

// ---------------------------------------------------------------------------
// Kernel 1: sequential forward LSTM cell. One block per batch element (the
// recurrence is independent across b). Thread tid owns hidden index h = tid.
// ---------------------------------------------------------------------------
__global__ __launch_bounds__(256) void eprop_fwd(
    const float* __restrict__ x,    const float* __restrict__ w_ih,
    const float* __restrict__ w_hh, const float* __restrict__ b_ih,
    const float* __restrict__ b_hh, const float* __restrict__ h0,
    const float* __restrict__ c0,   const float* __restrict__ gy,
    float* __restrict__ hseq,
    float* __restrict__ Mi, float* __restrict__ Mf, float* __restrict__ Mg,
    float* __restrict__ Fg, float* __restrict__ LpA, float* __restrict__ LoA,
    float* __restrict__ Hprev)
{
    __shared__ float xs[I_];
    __shared__ float hs[H_];
    const int b   = blockIdx.x;
    const int tid = threadIdx.x;          // == h index

    float c = c0[b * H_ + tid];
    hs[tid] = h0[b * H_ + tid];
    __syncthreads();

    for (int t = 0; t < S_; ++t) {
        xs[tid] = x[(t * B_ + b) * I_ + tid];
        __syncthreads();

        float pre[4];
        #pragma unroll
        for (int q = 0; q < 4; ++q) {
            const int r = q * H_ + tid;
            float acc = b_ih[r] + b_hh[r];
            const float* wi = w_ih + (size_t)r * I_;
            const float* wh = w_hh + (size_t)r * H_;
            #pragma unroll 4
            for (int i = 0; i < I_; ++i) acc += xs[i] * wi[i];
            #pragma unroll 4
            for (int j = 0; j < H_; ++j) acc += hs[j] * wh[j];
            pre[q] = acc;
        }

        const float ig = sigmoidf_(pre[0]);
        const float fg = sigmoidf_(pre[1]);
        const float gg = tanhf(pre[2]);
        const float og = sigmoidf_(pre[3]);

        const float hold = hs[tid];
        const float cn   = fg * c + ig * gg;
        const float tcn  = tanhf(cn);
        const float hn   = og * tcn;

        const float d_i = gg * ig * (1.0f - ig);
        const float d_f = c  * fg * (1.0f - fg);
        const float d_g = ig * (1.0f - gg * gg);

        const float Lt  = gy[(t * B_ + b) * H_ + tid];
        const float psi = og * (1.0f - tcn * tcn);
        const float Lp  = Lt * psi;
        const float Lo  = Lt * tcn * og * (1.0f - og);

        const int idx = (t * B_ + b) * H_ + tid;
        Mi[idx] = d_i;  Mf[idx] = d_f;  Mg[idx] = d_g;
        Fg[idx] = fg;   LpA[idx] = Lp;  LoA[idx] = Lo;
        Hprev[idx] = hold;
        hseq[idx]  = hn;

        c = cn;
        __syncthreads();
        hs[tid] = hn;
        __syncthreads();
    }
}

// ---------------------------------------------------------------------------
// Kernel 2: backward scan  L~[t] = Lp[t] + fg[t+1] * L~[t+1]  per (b,h),
// fused with M_gate[t] = d_gate[t] * L~[t] written in place.
// ---------------------------------------------------------------------------
__global__ __launch_bounds__(256) void eprop_bwd_scan(
    float* __restrict__ Mi, float* __restrict__ Mf, float* __restrict__ Mg,
    const float* __restrict__ Fg, const float* __restrict__ Lp)
{
    const int b = blockIdx.x;
    const int h = threadIdx.x;
    float lt = 0.0f, fgn = 0.0f;
    for (int t = S_ - 1; t >= 0; --t) {
        const int idx = (t * B_ + b) * H_ + h;
        lt  = Lp[idx] + fgn * lt;
        fgn = Fg[idx];
        Mi[idx] *= lt;
        Mf[idx] *= lt;
        Mg[idx] *= lt;
    }
}

// ---------------------------------------------------------------------------
// Async stage of one K-chunk: global [k][col] tiles -> LDS [col][k] tiles.
// Per-lane LDS destination address performs the transpose scatter for free;
// global side stays coalesced. 24 async issues per wave, tracked by ASYNCcnt.
// ---------------------------------------------------------------------------
__device__ __forceinline__ void stage_async(
    unsigned at_base, unsigned bt_base,
    unsigned long long a_sbase, unsigned long long b_sbase,
    int k0, int am, int aks, int bnn, int bks, int acol0, int nbase)
{
    #pragma unroll
    for (int p = 0; p < 16; ++p) {
        const int k = p * 2 + aks;
        const unsigned lds = at_base + (unsigned)((am * PAD + k) * 4);
        const unsigned go  = (unsigned)(((k0 + k) * H_ + acol0 + am) * 4);
        asm volatile("global_load_async_to_lds_b32 %0, %1, %2"
                     :: "v"(lds), "v"(go), "s"(a_sbase) : "memory");
    }
    #pragma unroll
    for (int p = 0; p < 8; ++p) {
        const int k = p * 4 + bks;
        const unsigned lds = bt_base + (unsigned)((bnn * PAD + k) * 4);
        const unsigned go  = (unsigned)(((k0 + k) * I_ + nbase + bnn) * 4);
        asm volatile("global_load_async_to_lds_b32 %0, %1, %2"
                     :: "v"(lds), "v"(go), "s"(b_sbase) : "memory");
    }
}

// ---------------------------------------------------------------------------
// Kernel 3: gW[g*H + h, n] = sum_k M_g[k, h] * Bsrc[k, n],  K = 1024.
// Double-buffered, async-staged, LDS-tiled FP32 WMMA 16x16x4:
//   - Block tile 128(M) x 64(N); K in 32-wide chunks, 2 LDS buffers (52 KB).
//   - While chunk n is consumed by v_wmma, chunk n+1 streams in via
//     GLOBAL_LOAD_ASYNC_TO_LDS_B32. Async loads complete in order, so
//     `s_wait_asynccnt 24` (= the prefetch still in flight) guarantees
//     chunk n has fully landed without draining the pipe.
//   - LDS tiles are k-transposed with padded rows: each A/B fragment is one
//     8-byte ds_load (compiler pairs them into ds_load_2addr_b64).
//   - Each wave: 2x2 grid of 16x16 f32 accumulators, 4 v_wmma per k-step.
// ---------------------------------------------------------------------------
__global__ __launch_bounds__(256) void eprop_gemm_wmma(
    const float* __restrict__ Mi, const float* __restrict__ Mf,
    const float* __restrict__ Mg, const float* __restrict__ Mo,
    const float* __restrict__ Bsrc, float* __restrict__ out /* 1024 x 256 */)
{
    __shared__ float At[2][BM * PAD];   // [buf][m][k] transposed A tiles
    __shared__ float Bt[2][BN * PAD];   // [buf][n][k] transposed B tiles

    const int bm    = blockIdx.x >> 2;      // 0..7  (128-row slabs of 4H)
    const int bn    = blockIdx.x & 3;       // 0..3  (64-col slabs)
    const int hbase = bm * BM;              // output row base (within 4H)
    const int gate  = bm >> 1;              // 0..3
    const int acol0 = (bm & 1) * BM;        // column base within gate's (K x 256) A
    const int nbase = bn * BN;

    const float* A = (gate == 0) ? Mi : (gate == 1) ? Mf : (gate == 2) ? Mg : Mo;

    const int tid  = threadIdx.x;
    const int lane = tid & 31;
    const int w    = tid >> 5;              // wave 0..7
    const int wm   = w >> 1;                // 0..3 -> 32-row strip
    const int wn   = w & 1;                 // 0..1 -> 32-col strip
    const int hl   = lane & 15;
    const int ko   = (lane < 16) ? 0 : 2;   // f32 A/B frag k-offset per half-wave

    // staging decomposition (256 threads)
    const int am  = tid & 127;              // A: m index, contiguous -> coalesced
    const int aks = tid >> 7;               // A: k sub-index 0..1
    const int bnn = tid & 63;               // B: n index, contiguous -> coalesced
    const int bks = tid >> 6;               // B: k sub-index 0..3

    const unsigned at_base0 = (unsigned)(size_t)&At[0][0]; // LDS aperture low bits
    const unsigned at_base1 = (unsigned)(size_t)&At[1][0];
    const unsigned bt_base0 = (unsigned)(size_t)&Bt[0][0];
    const unsigned bt_base1 = (unsigned)(size_t)&Bt[1][0];
    const unsigned long long a_sbase = (unsigned long long)(size_t)A;
    const unsigned long long b_sbase = (unsigned long long)(size_t)Bsrc;

    v8f acc[2][2];
    #pragma unroll
    for (int i = 0; i < 2; ++i)
        #pragma unroll
        for (int j = 0; j < 2; ++j)
            acc[i][j] = (v8f){0.f,0.f,0.f,0.f,0.f,0.f,0.f,0.f};

    // prologue: stage chunk 0 into buffer 0
    stage_async(at_base0, bt_base0, a_sbase, b_sbase,
                0, am, aks, bnn, bks, acol0, nbase);

    for (int ch = 0; ch < NCH; ++ch) {
        const int cur = ch & 1;
        if (ch + 1 < NCH) {
            // prefetch next chunk into the other buffer, then wait only for
            // the *previous* 24 async issues (in-order completion).
            stage_async(cur ? at_base0 : at_base1, cur ? bt_base0 : bt_base1,
                        a_sbase, b_sbase,
                        (ch + 1) * KC, am, aks, bnn, bks, acol0, nbase);
            asm volatile("s_wait_asynccnt 0x18" ::: "memory"); // <= 24 in flight
        } else {
            asm volatile("s_wait_asynccnt 0x0" ::: "memory");
        }
        __syncthreads();

        // ---- WMMA over the staged chunk ----
        #pragma unroll
        for (int kk = 0; kk < KC; kk += 4) {
            const v2f a0 = *(const v2f*)&At[cur][(wm * 32 +      hl) * PAD + kk + ko];
            const v2f a1 = *(const v2f*)&At[cur][(wm * 32 + 16 + hl) * PAD + kk + ko];
            const v2f b0 = *(const v2f*)&Bt[cur][(wn * 32 +      hl) * PAD + kk + ko];
            const v2f b1 = *(const v2f*)&Bt[cur][(wn * 32 + 16 + hl) * PAD + kk + ko];
            acc[0][0] = __builtin_amdgcn_wmma_f32_16x16x4_f32(
                false, a0, false, b0, (short)0, acc[0][0], false, false);
            acc[0][1] = __builtin_amdgcn_wmma_f32_16x16x4_f32(
                false, a0, false, b1, (short)0, acc[0][1], false, false);
            acc[1][0] = __builtin_amdgcn_wmma_f32_16x16x4_f32(
                false, a1, false, b0, (short)0, acc[1][0], false, false);
            acc[1][1] = __builtin_amdgcn_wmma_f32_16x16x4_f32(
                false, a1, false, b1, (short)0, acc[1][1], false, false);
        }
        __syncthreads();   // buffer `cur` free for the chunk-(ch+2) prefetch
    }

    // ---- store: D vgpr r -> M = r (+8 for lanes 16..31), N = lane%16 ----
    const int rbase = (lane < 16) ? 0 : 8;
    #pragma unroll
    for (int i = 0; i < 2; ++i) {
        #pragma unroll
        for (int j = 0; j < 2; ++j) {
            const int mrow0 = hbase + wm * 32 + i * 16 + rbase;
            const int ncol  = nbase + wn * 32 + j * 16 + hl;
            #pragma unroll
            for (int r = 0; r < 8; ++r)
                out[(size_t)(mrow0 + r) * I_ + ncol] = acc[i][j][r];
        }
    }
}

// ---------------------------------------------------------------------------
// Kernel 4: gb[g*H + h] = sum_k M_g[k, h]
// ---------------------------------------------------------------------------
__global__ __launch_bounds__(256) void eprop_gbias(
    const float* __restrict__ Mi, const float* __restrict__ Mf,
    const float* __restrict__ Mg, const float* __restrict__ Mo,
    float* __restrict__ gb)
{
    const int id   = blockIdx.x * 256 + threadIdx.x;   // 0..1023
    const int gate = id >> 8;
    const int h    = id & 255;
    const float* A = (gate == 0) ? Mi : (gate == 1) ? Mf : (gate == 2) ? Mg : Mo;
    float s = 0.0f;
    for (int k = 0; k < K_; ++k) s += A[(size_t)k * H_ + h];
    gb[id] = s;
}

// ---------------------------------------------------------------------------
extern "C" void kernel_launch(void* const* d_in, const int* in_sizes, int n_in,
                              void* d_out, int out_size, void* d_ws, size_t ws_size,
                              hipStream_t stream)
{
    const float* x    = (const float*)d_in[0];   // (S,B,I)
    const float* w_ih = (const float*)d_in[1];   // (4H,I)
    const float* w_hh = (const float*)d_in[2];   // (4H,H)
    const float* b_ih = (const float*)d_in[3];   // (4H)
    const float* b_hh = (const float*)d_in[4];   // (4H)
    const float* h0   = (const float*)d_in[5];   // (B,H)
    const float* c0   = (const float*)d_in[6];   // (B,H)
    const float* gy   = (const float*)d_in[7];   // (S,B,H)

    float* ws   = (float*)d_ws;
    float* Mi   = ws + 0 * (size_t)SBH;   // d_i  -> L~*d_i
    float* Mf   = ws + 1 * (size_t)SBH;   // d_f  -> L~*d_f
    float* Mg   = ws + 2 * (size_t)SBH;   // d_g  -> L~*d_g
    float* Fg   = ws + 3 * (size_t)SBH;
    float* Lp   = ws + 4 * (size_t)SBH;
    float* Lo   = ws + 5 * (size_t)SBH;   // == Mo
    float* Hpr  = ws + 6 * (size_t)SBH;

    float* hseq = (float*)d_out;               // (S,B,H)
    float* gWih = hseq + (size_t)SBH;          // (4H,I)
    float* gWhh = gWih + (size_t)H4_ * I_;     // (4H,H)
    float* gb   = gWhh + (size_t)H4_ * H_;     // (4H)

    eprop_fwd<<<B_, 256, 0, stream>>>(x, w_ih, w_hh, b_ih, b_hh, h0, c0, gy,
                                      hseq, Mi, Mf, Mg, Fg, Lp, Lo, Hpr);
    eprop_bwd_scan<<<B_, 256, 0, stream>>>(Mi, Mf, Mg, Fg, Lp);
    // 32 blocks per weight matrix: (1024/128) x (256/64)
    eprop_gemm_wmma<<<32, 256, 0, stream>>>(Mi, Mf, Mg, Lo, x,   gWih);
    eprop_gemm_wmma<<<32, 256, 0, stream>>>(Mi, Mf, Mg, Lo, Hpr, gWhh);
    eprop_gbias<<<4, 256, 0, stream>>>(Mi, Mf, Mg, Lo, gb);
}